// DenseGATLayer_47717086659080
// MI455X (gfx1250) — compile-verified
//
#include <hip/hip_runtime.h>
#include <hip/hip_bf16.h>
#include <math.h>

typedef __attribute__((ext_vector_type(2))) float v2f;
typedef __attribute__((ext_vector_type(8))) float v8f;

#define N     8192
#define F_IN  128
#define F_OUT 64
#define SLOPE 0.2f

// ---------------------------------------------------------------------------
// Kernel 1: Wh = h @ W^T  via V_WMMA_F32_16X16X4_F32
// block = 128 threads (4 waves). Wave w computes the 16x16 tile at
// rows [blockIdx.x*16, +16), cols [w*16, +16). K-loop over 128 in steps of 4.
// A fragment (16x4 f32): lane<16 -> M=lane, K={0,1}; lane>=16 -> M=lane-16, K={2,3}
// B fragment (4x16 f32): lane<16 -> N=lane, K={0,1}; lane>=16 -> N=lane-16, K={2,3}
// B[k][n] = W^T[k][n] = W[n][k]
// ---------------------------------------------------------------------------
__global__ void __launch_bounds__(128) gat_wh_wmma(const float* __restrict__ h,
                                                   const float* __restrict__ W,
                                                   float* __restrict__ Wh) {
  const int lane = threadIdx.x & 31;
  const int wave = threadIdx.x >> 5;          // col tile 0..3
  const int r0   = blockIdx.x * 16;
  const int n0   = wave * 16;
  const int m    = lane & 15;
  const int kk   = (lane >> 4) << 1;          // 0 or 2

  v8f acc = {};
  const float* hrow = h + (size_t)(r0 + m) * F_IN + kk;
  const float* wrow = W + (size_t)(n0 + m) * F_IN + kk;
  for (int k = 0; k < F_IN; k += 4) {
    v2f a, b;
    a.x = hrow[k];     a.y = hrow[k + 1];
    b.x = wrow[k];     b.y = wrow[k + 1];
    acc = __builtin_amdgcn_wmma_f32_16x16x4_f32(false, a, false, b,
                                                (short)0, acc, false, false);
  }
  // D layout: VGPR r -> M = r + 8*(lane>=16), N = lane&15
  const int mbase = r0 + ((lane >> 4) << 3);
#pragma unroll
  for (int r = 0; r < 8; ++r)
    Wh[(size_t)(mbase + r) * F_OUT + n0 + m] = acc[r];
}

// ---------------------------------------------------------------------------
// Kernel 2: e_left[i] = Wh[i,:]·a_left ; e_right[i] = Wh[i,:]·a_right
// ---------------------------------------------------------------------------
__global__ void __launch_bounds__(256) gat_evec(const float* __restrict__ Wh,
                                                const float* __restrict__ a_left,
                                                const float* __restrict__ a_right,
                                                float* __restrict__ e_left,
                                                float* __restrict__ e_right) {
  const int i = blockIdx.x * blockDim.x + threadIdx.x;
  if (i >= N) return;
  const float4* row = (const float4*)(Wh + (size_t)i * F_OUT);
  const float4* al4 = (const float4*)a_left;
  const float4* ar4 = (const float4*)a_right;
  float sl = 0.f, sr = 0.f;
#pragma unroll
  for (int q = 0; q < F_OUT / 4; ++q) {
    float4 v  = row[q];
    float4 al = al4[q];
    float4 ar = ar4[q];
    sl += v.x * al.x + v.y * al.y + v.z * al.z + v.w * al.w;
    sr += v.x * ar.x + v.y * ar.y + v.z * ar.z + v.w * ar.w;
  }
  e_left[i]  = sl;
  e_right[i] = sr;
}

// ---------------------------------------------------------------------------
// Kernel 3: per-row softmax stats over nonzero A entries (A = adj + I).
// One wave per row; coalesced stream of the 32KB row; per-lane online
// (max, sum-exp); shfl_xor tree combine.
// ---------------------------------------------------------------------------
__global__ void __launch_bounds__(256) gat_rowstats(const float* __restrict__ adj,
                                                    const float* __restrict__ e_left,
                                                    const float* __restrict__ e_right,
                                                    float* __restrict__ row_max,
                                                    float* __restrict__ row_inv) {
  const int lane = threadIdx.x & 31;
  const int row  = blockIdx.x * 8 + (threadIdx.x >> 5);
  const float el = e_left[row];
  const float* arow = adj + (size_t)row * N;

  float mloc = -INFINITY, sloc = 0.f;
  for (int j = lane; j < N; j += 32) {
    float a = arow[j];
    if (j == row) a += 1.0f;                 // self-loop: A = adj + I
    if (a != 0.f) {
      float e = el + e_right[j];
      e = (e >= 0.f) ? e : SLOPE * e;        // LeakyReLU(0.2)
      float mn = fmaxf(mloc, e);
      sloc = sloc * __expf(mloc - mn) + __expf(e - mn);
      mloc = mn;
    }
  }
  // tree reduce (m, s) pairs across 32 lanes
#pragma unroll
  for (int off = 16; off >= 1; off >>= 1) {
    float m2 = __shfl_xor(mloc, off, 32);
    float s2 = __shfl_xor(sloc, off, 32);
    float mn = fmaxf(mloc, m2);
    if (mn == -INFINITY) {
      sloc = 0.f;                            // both halves empty (never final)
    } else {
      sloc = sloc * __expf(mloc - mn) + s2 * __expf(m2 - mn);
    }
    mloc = mn;
  }
  if (lane == 0) {
    row_max[row] = mloc;
    row_inv[row] = 1.0f / sloc;              // diagonal guarantees sloc > 0
  }
}

// ---------------------------------------------------------------------------
// Kernel 4: out = attention @ Wh, attention rebuilt on the fly.
// One wave per 16-row tile; 4 WMMA f32 accumulators = 16x64 out tile.
// K-loop over all 8192 columns in chunks of 4.
// ---------------------------------------------------------------------------
__global__ void __launch_bounds__(128) gat_attn_out_wmma(const float* __restrict__ adj,
                                                         const float* __restrict__ Wh,
                                                         const float* __restrict__ e_left,
                                                         const float* __restrict__ e_right,
                                                         const float* __restrict__ row_max,
                                                         const float* __restrict__ row_inv,
                                                         float* __restrict__ out) {
  const int lane = threadIdx.x & 31;
  const int wave = threadIdx.x >> 5;
  const int r0   = (blockIdx.x * 4 + wave) * 16;
  const int m    = lane & 15;
  const int kk   = (lane >> 4) << 1;         // 0 or 2

  const int   row = r0 + m;
  const float el  = e_left[row];
  const float mx  = row_max[row];
  const float inv = row_inv[row];
  const float* arow = adj + (size_t)row * N;

  v8f acc0 = {}, acc1 = {}, acc2 = {}, acc3 = {};

  for (int j = 0; j < N; j += 4) {
    if ((j & 255) == 0)                       // keep the adj stream ahead
      __builtin_prefetch(arow + j + 512, 0, 0);

    const int c0 = j + kk, c1 = c0 + 1;
    float a0 = arow[c0]; if (c0 == row) a0 += 1.0f;
    float a1 = arow[c1]; if (c1 == row) a1 += 1.0f;

    float w0 = 0.f, w1 = 0.f;
    if (a0 != 0.f) {
      float e = el + e_right[c0];
      e = (e >= 0.f) ? e : SLOPE * e;
      w0 = __expf(e - mx) * inv * a0;         // softmax * A (valued mask)
    }
    if (a1 != 0.f) {
      float e = el + e_right[c1];
      e = (e >= 0.f) ? e : SLOPE * e;
      w1 = __expf(e - mx) * inv * a1;
    }
    v2f afr; afr.x = w0; afr.y = w1;

    const float* b0p = Wh + (size_t)c0 * F_OUT;   // B[k][n] = Wh[j+k][n]
    const float* b1p = Wh + (size_t)c1 * F_OUT;
    v2f b;
    b.x = b0p[m];       b.y = b1p[m];
    acc0 = __builtin_amdgcn_wmma_f32_16x16x4_f32(false, afr, false, b, (short)0, acc0, false, false);
    b.x = b0p[16 + m];  b.y = b1p[16 + m];
    acc1 = __builtin_amdgcn_wmma_f32_16x16x4_f32(false, afr, false, b, (short)0, acc1, false, false);
    b.x = b0p[32 + m];  b.y = b1p[32 + m];
    acc2 = __builtin_amdgcn_wmma_f32_16x16x4_f32(false, afr, false, b, (short)0, acc2, false, false);
    b.x = b0p[48 + m];  b.y = b1p[48 + m];
    acc3 = __builtin_amdgcn_wmma_f32_16x16x4_f32(false, afr, false, b, (short)0, acc3, false, false);
  }

  const int mbase = r0 + ((lane >> 4) << 3);
#pragma unroll
  for (int r = 0; r < 8; ++r) {
    float* orow = out + (size_t)(mbase + r) * F_OUT + m;
    orow[0]  = acc0[r];
    orow[16] = acc1[r];
    orow[32] = acc2[r];
    orow[48] = acc3[r];
  }
}

// ---------------------------------------------------------------------------
extern "C" void kernel_launch(void* const* d_in, const int* in_sizes, int n_in,
                              void* d_out, int out_size, void* d_ws, size_t ws_size,
                              hipStream_t stream) {
  const float* h       = (const float*)d_in[0];   // [8192,128]
  const float* W       = (const float*)d_in[1];   // [64,128]
  const float* a_left  = (const float*)d_in[2];   // [64]
  const float* a_right = (const float*)d_in[3];   // [64]
  const float* adj     = (const float*)d_in[4];   // [8192,8192]
  float* out = (float*)d_out;                     // [8192,64]

  char* ws = (char*)d_ws;
  float* Wh      = (float*)(ws);                              // 2 MB
  float* e_left  = (float*)(ws + (size_t)N * F_OUT * 4);      // 32 KB
  float* e_right = e_left  + N;
  float* row_max = e_right + N;
  float* row_inv = row_max + N;

  // 1) Wh = h @ W^T   (512 blocks x 4 waves, one 16x16 tile per wave)
  gat_wh_wmma<<<N / 16, 128, 0, stream>>>(h, W, Wh);

  // 2) attention score vectors
  gat_evec<<<N / 256, 256, 0, stream>>>(Wh, a_left, a_right, e_left, e_right);

  // 3) per-row softmax max / inverse-sum (one wave per row)
  gat_rowstats<<<N / 8, 256, 0, stream>>>(adj, e_left, e_right, row_max, row_inv);

  // 4) out = softmax(masked e) * A @ Wh   (one wave per 16-row tile)
  gat_attn_out_wmma<<<N / 64, 128, 0, stream>>>(adj, Wh, e_left, e_right,
                                                row_max, row_inv, out);
}